// MultiVariateGaussian_50861002719709
// MI455X (gfx1250) — compile-verified
//
#include <hip/hip_runtime.h>

// Problem constants (from reference): B=256, C=100, H=W=56 -> P=3136
#define BATCH 256
#define CDIM  100
#define PDIM  3136          // 56*56
#define CPAD  112           // 7 WMMA tiles of 16
#define XSTR  260           // LDS row stride for X [CPAD][BATCH], padded vs 256
#define AS    204           // LDS row stride for augmented [100 x 200] matrix
#define NT    7             // 7x7 tile grid; 28 upper-triangle tiles computed
#define EPS_DIAG 0.01f

typedef __attribute__((ext_vector_type(2))) float v2f;
typedef __attribute__((ext_vector_type(8))) float v8f;
typedef __attribute__((ext_vector_type(4))) int   v4i;

// pointer-to-int4 in global (AS1) and LDS (AS3) address spaces
typedef v4i __attribute__((address_space(1)))* gas_v4i_ptr;
typedef v4i __attribute__((address_space(3)))* las_v4i_ptr;

#if defined(__has_builtin)
#if __has_builtin(__builtin_amdgcn_global_load_async_to_lds_b128)
#define HAVE_ASYNC_LDS 1
#endif
#endif

// ---------------------------------------------------------------------------
// Kernel 1: transpose embedding [B][C][P] -> ws [P][C][B] so each patch's
// [C][B] slice is contiguous. Tiled 32x32 (b x p) per fixed c through LDS.
// grid = (btile=8, ptile=98, c=100), block = 256 (32x8)
// ---------------------------------------------------------------------------
__global__ __launch_bounds__(256) void transpose_bcp_to_pcb(const float* __restrict__ emb,
                                                            float* __restrict__ ws) {
    __shared__ float tile[32][33];
    const int tx = threadIdx.x & 31;
    const int ty = threadIdx.x >> 5;           // 0..7
    const int btile = blockIdx.x;              // 0..7
    const int ptile = blockIdx.y;              // 0..97
    const int c     = blockIdx.z;              // 0..99

    #pragma unroll
    for (int i = 0; i < 4; ++i) {
        int row = ty + i * 8;                   // b within tile
        int b   = btile * 32 + row;
        size_t src = ((size_t)(b * CDIM + c)) * PDIM + (size_t)ptile * 32 + tx;
        tile[row][tx] = emb[src];
    }
    __syncthreads();

    #pragma unroll
    for (int i = 0; i < 4; ++i) {
        int pr = ty + i * 8;                    // p within tile
        size_t p = (size_t)ptile * 32 + pr;
        size_t dst = p * (CDIM * BATCH) + (size_t)c * BATCH + btile * 32 + tx;
        ws[dst] = tile[tx][pr];
    }
}

// ---------------------------------------------------------------------------
// Kernel 2: per patch p (one workgroup, 8 waves):
//   1) fill LDS with the contiguous [100][256] slice via async-to-LDS b128
//   2) mean over batch -> write mean output, center rows, zero pad rows
//   3) cov = X*X^T / 255 via V_WMMA_F32_16X16X4_F32: 28 upper-triangle tiles,
//      results written straight to the (disjoint) Aug LDS region + mirrored
//   4) Gauss-Jordan inverse of (cov + 0.01*I) on augmented [100x200] in LDS
//   5) coalesced store of the inverse
// Dynamic LDS: 112*260 + 100*204 floats = 198,080 B (1 WG per WGP).
// ---------------------------------------------------------------------------
__global__ __launch_bounds__(256) void cov_inv_kernel(const float* __restrict__ ws,
                                                      float* __restrict__ out) {
    extern __shared__ float smem[];
    float* X   = smem;                       // [CPAD][XSTR]
    float* Aug = smem + CPAD * XSTR;         // [CDIM][AS], disjoint from X
    __shared__ float meanArr[CDIM];
    __shared__ float facArr[CDIM];

    const int tid  = threadIdx.x;
    const int lane = tid & 31;
    const int w    = tid >> 5;         // wave id 0..7
    const int p    = blockIdx.x;       // patch id

    // ---- 1) load slice: 100*64 x 16B ----
    const float* base = ws + (size_t)p * (CDIM * BATCH);
#ifdef HAVE_ASYNC_LDS
    for (int i = tid; i < CDIM * (BATCH / 4); i += 256) {
        int c = i >> 6;                // /64
        int q = i & 63;
        __builtin_amdgcn_global_load_async_to_lds_b128(
            (gas_v4i_ptr)(base + c * BATCH + q * 4),
            (las_v4i_ptr)(X + c * XSTR + q * 4),
            0, 0);
    }
#if __has_builtin(__builtin_amdgcn_s_wait_asynccnt)
    __builtin_amdgcn_s_wait_asynccnt(0);
#else
    asm volatile("s_wait_asynccnt 0x0" ::: "memory");
#endif
#else
    for (int i = tid; i < CDIM * (BATCH / 4); i += 256) {
        int c = i >> 6;
        int q = i & 63;
        *(float4*)(X + c * XSTR + q * 4) = *(const float4*)(base + c * BATCH + q * 4);
    }
#endif
    __syncthreads();

    // ---- 2) mean + center + zero-pad (identity init runs here too: Aug is disjoint) ----
    if (tid < CDIM) {
        const float* row = X + tid * XSTR;
        float s = 0.f;
        for (int b = 0; b < BATCH; ++b) s += row[b];
        float mn = s * (1.0f / BATCH);
        meanArr[tid] = mn;
        out[(size_t)tid * PDIM + p] = mn;          // mean output [C][P]
    }
    for (int idx = tid; idx < CDIM * CDIM; idx += 256) {
        int r = idx / CDIM, j = idx - r * CDIM;
        Aug[r * AS + CDIM + j] = (r == j) ? 1.f : 0.f;
    }
    __syncthreads();
    for (int idx = tid; idx < CPAD * BATCH; idx += 256) {
        int c = idx >> 8;
        int b = idx & 255;
        float* ptr = X + c * XSTR + b;
        *ptr = (c < CDIM) ? (*ptr - meanArr[c]) : 0.f;
    }
    __syncthreads();

    // ---- 3) covariance via fp32 WMMA: 28 upper-triangle tiles over 8 waves ----
    // A fragment (16x4): lane<16 holds row M=lane, K={k,k+1}; lane>=16 -> K={k+2,k+3}
    // B fragment (4x16) of X^T has identical addressing with the tj row base.
    const int koff = (lane >> 4) << 1;             // 0 or 2
    for (int u = w; u < (NT * (NT + 1)) / 2; u += 8) {
        // decompose u -> (ti, tj) with ti <= tj (wave-uniform scalar loop)
        int ti = 0, rem = u;
        while (rem >= NT - ti) { rem -= NT - ti; ++ti; }
        int tj = ti + rem;

        const float* arow = X + (16 * ti + (lane & 15)) * XSTR + koff;
        const float* brow = X + (16 * tj + (lane & 15)) * XSTR + koff;
        v8f acc = {0.f, 0.f, 0.f, 0.f, 0.f, 0.f, 0.f, 0.f};
        #pragma unroll 8
        for (int k = 0; k < BATCH; k += 4) {
            v2f a = *(const v2f*)(arow + k);
            v2f b = *(const v2f*)(brow + k);
            acc = __builtin_amdgcn_wmma_f32_16x16x4_f32(
                false, a, false, b, (short)0, acc, false, false);
        }
        // write tile (and mirror) directly into Aug; no cross-barrier registers
        int n = 16 * tj + (lane & 15);
        int mbase = 16 * ti + ((lane >> 4) << 3);
        if (n < CDIM) {
            #pragma unroll
            for (int v = 0; v < 8; ++v) {
                int m = mbase + v;
                if (m < CDIM) {
                    float val = acc[v] * (1.0f / (BATCH - 1));
                    Aug[m * AS + n] = (m == n) ? (val + EPS_DIAG) : val;
                    if (ti != tj) Aug[n * AS + m] = val;   // symmetric mirror
                }
            }
        }
    }
    __syncthreads();

    // ---- 4) Gauss-Jordan (SPD + eps*I => no pivoting) ----
    for (int k = 0; k < CDIM; ++k) {
        if (tid < CDIM) facArr[tid] = Aug[tid * AS + k];
        __syncthreads();
        float pv = 1.0f / facArr[k];
        float pr = 0.f;
        if (tid < 2 * CDIM) {
            pr = Aug[k * AS + tid] * pv;
            Aug[k * AS + tid] = pr;
        }
        __syncthreads();
        if (tid < 2 * CDIM) {
            for (int r = 0; r < CDIM; ++r) {
                if (r == k) continue;
                Aug[r * AS + tid] -= facArr[r] * pr;
            }
        }
        __syncthreads();
    }

    // ---- 5) store inverse [P][C][C], coalesced ----
    float* outInv = out + (size_t)CDIM * PDIM + (size_t)p * (CDIM * CDIM);
    for (int idx = tid; idx < CDIM * CDIM; idx += 256) {
        int r = idx / CDIM, j = idx - r * CDIM;
        outInv[idx] = Aug[r * AS + CDIM + j];
    }
}

// ---------------------------------------------------------------------------
// kernel_launch
//   d_in[0]: embedding f32 [256,100,56,56]
//   d_out  : mean [100,3136] ++ inv_cov [3136,100,100]  (f32)
//   d_ws   : needs 3136*100*256*4 = 321,126,400 bytes for the transposed copy
// ---------------------------------------------------------------------------
extern "C" void kernel_launch(void* const* d_in, const int* in_sizes, int n_in,
                              void* d_out, int out_size, void* d_ws, size_t ws_size,
                              hipStream_t stream) {
    const float* emb = (const float*)d_in[0];
    float* out = (float*)d_out;
    float* ws  = (float*)d_ws;

    dim3 g1(BATCH / 32, PDIM / 32, CDIM);      // (8, 98, 100)
    transpose_bcp_to_pcb<<<g1, dim3(256), 0, stream>>>(emb, ws);

    size_t lds_bytes = (size_t)(CPAD * XSTR + CDIM * AS) * sizeof(float); // 198,080 B
    cov_inv_kernel<<<dim3(PDIM), dim3(256), lds_bytes, stream>>>(ws, out);
}